// DenseEquivariantPointAttention_88158498718348
// MI455X (gfx1250) — compile-verified
//
#include <hip/hip_runtime.h>
#include <math.h>

// ---------------------------------------------------------------------------
// Problem constants
// ---------------------------------------------------------------------------
#define NRES 768
#define CS   384
#define CZ   128
#define NH   12
#define CH   64
#define PQ   4
#define PV   8
#define N0C  192
#define N1C  64
#define CZ4  32

typedef float v8f __attribute__((ext_vector_type(8)));
typedef float v2f __attribute__((ext_vector_type(2)));

// ---------------------------------------------------------------------------
// Generic strided / batched fp32 WMMA GEMM:  C[m,n] (+)= sum_k A[m,k]*B[k,n] (+bias[n])
// A elem (m,k) = A[m*ams + k*aks], B elem (k,n) = B[k*bks + n*bns],
// C elem (m,n) = C[m*cms + n*cns].  blockIdx.z selects batch via *bat strides.
//
// One wave computes a 16x32 tile: two V_WMMA_F32_16X16X4_F32 accumulators share
// one A fragment (halves A traffic, doubles WMMA RAW distance).
// Lane layout per CDNA5 ISA 7.12.2 (f32, K-step 4):
//   A frag (16x4):  lane<16 -> M=lane,K={k,k+1}; lane>=16 -> M=lane-16,K={k+2,k+3}
//   B frag (4x16):  lane<16 -> N=lane,K={k,k+1}; lane>=16 -> N=lane-16,K={k+2,k+3}
//   C frag: vgpr r -> M = r + 8*(lane>>4), N = lane&15
//
// Out-of-range handling: addresses are CLAMPED to valid rows/cols (branchless
// loads, EXEC stays all-ones); garbage only lands in accumulator elements whose
// stores are guarded, so no value masking is needed.  Requires K % 4 == 0.
// AK1/BK1 specialize unit K-stride into 8-byte vector loads.
// ---------------------------------------------------------------------------
template <bool AK1, bool BK1>
__global__ void wmma_gemm_f32(const float* __restrict__ A,
                              const float* __restrict__ B,
                              const float* __restrict__ bias,
                              float* __restrict__ C,
                              int M, int N, int K,
                              long ams, long aks, long bks, long bns,
                              long cms, long cns,
                              long abat, long bbat, long cbat,
                              int beta, int waves_n)
{
    A += (long)blockIdx.z * abat;
    B += (long)blockIdx.z * bbat;
    C += (long)blockIdx.z * cbat;

    const int wave = threadIdx.x >> 5;
    const int lane = threadIdx.x & 31;
    const int hlf  = lane >> 4;      // 0 or 1
    const int l16  = lane & 15;
    const int wm   = 8 / waves_n;
    const int wrow = wave / waves_n;
    const int wcol = wave % waves_n;
    const int tile_m = (blockIdx.x * wm + wrow) * 16;
    const int tile_n = (blockIdx.y * waves_n + wcol) * 32;

    const int mrow0 = tile_m + hlf * 8;
    const int n0    = tile_n + l16;
    const int n1    = n0 + 16;
    const int n0c   = (n0 < N) ? n0 : 0;
    const int n1c   = (n1 < N) ? n1 : 0;

    v8f acc0, acc1;
    if (beta) {
        for (int r = 0; r < 8; ++r) {
            int m  = mrow0 + r;
            int mc = (m < M) ? m : 0;
            acc0[r] = C[(long)mc * cms + (long)n0c * cns];
            acc1[r] = C[(long)mc * cms + (long)n1c * cns];
        }
    } else {
        for (int r = 0; r < 8; ++r) { acc0[r] = 0.f; acc1[r] = 0.f; }
    }

    const int am  = tile_m + l16;
    const int amc = (am < M) ? am : 0;

    const float* __restrict__ Ap  = A + (long)amc * ams + (long)(2 * hlf) * aks;
    const float* __restrict__ Bp0 = B + (long)n0c * bns + (long)(2 * hlf) * bks;
    const float* __restrict__ Bp1 = B + (long)n1c * bns + (long)(2 * hlf) * bks;
    const long astep = 4 * aks;
    const long bstep = 4 * bks;

    const int steps = K >> 2;
#pragma unroll 8
    for (int it = 0; it < steps; ++it) {
        v2f af, bf0, bf1;
        if (AK1) {
            af = *(const v2f*)Ap;
        } else {
            af.x = Ap[0]; af.y = Ap[aks];
        }
        if (BK1) {
            bf0 = *(const v2f*)Bp0;
            bf1 = *(const v2f*)Bp1;
        } else {
            bf0.x = Bp0[0]; bf0.y = Bp0[bks];
            bf1.x = Bp1[0]; bf1.y = Bp1[bks];
        }
        acc0 = __builtin_amdgcn_wmma_f32_16x16x4_f32(false, af, false, bf0,
                                                     (short)0, acc0, false, false);
        acc1 = __builtin_amdgcn_wmma_f32_16x16x4_f32(false, af, false, bf1,
                                                     (short)0, acc1, false, false);
        Ap += astep; Bp0 += bstep; Bp1 += bstep;
    }

    float bv0 = 0.f, bv1 = 0.f;
    if (bias) {
        if (n0 < N) bv0 = bias[n0];
        if (n1 < N) bv1 = bias[n1];
    }
    for (int r = 0; r < 8; ++r) {
        int m = mrow0 + r;
        if (m < M) {
            if (n0 < N) C[(long)m * cms + (long)n0 * cns] = acc0[r] + bv0;
            if (n1 < N) C[(long)m * cms + (long)n1 * cns] = acc1[r] + bv1;
        }
    }
}

// ---------------------------------------------------------------------------
// Build fused [wb | wdz] (128 x 44) weight + [bb | bdz] bias so z is streamed once.
// ---------------------------------------------------------------------------
__global__ void k_concat_w(const float* __restrict__ wb, const float* __restrict__ bb,
                           const float* __restrict__ wdz, const float* __restrict__ bdz,
                           float* __restrict__ WCAT, float* __restrict__ BCAT)
{
    int i = blockIdx.x * 256 + threadIdx.x;
    if (i < 128 * 44) {
        int r = i / 44, c = i % 44;
        WCAT[i] = (c < NH) ? wb[r * NH + c] : wdz[r * CZ4 + (c - NH)];
    }
    if (i < 44) BCAT[i] = (i < NH) ? bb[i] : bdz[i - NH];
}

// ---------------------------------------------------------------------------
// Select res-path vs atom-path q/k/v  (kv packed per head as [k(64) | v(64)])
// ---------------------------------------------------------------------------
__global__ void k_select_qkv(const unsigned char* __restrict__ is_atom,
                             const float* __restrict__ QA,
                             const float* __restrict__ KVR,
                             const float* __restrict__ KVA,
                             float* __restrict__ Q, float* __restrict__ Kb,
                             float* __restrict__ Vb)
{
    long i = (long)blockIdx.x * 256 + threadIdx.x;
    if (i >= (long)NRES * (NH * CH)) return;
    int n = (int)(i / (NH * CH)), j = (int)(i % (NH * CH));
    bool at = is_atom[n] != 0;
    Q[i] = at ? QA[i] : Q[i];
    int h = j >> 6, c = j & 63;
    const float* kv = at ? KVA : KVR;
    Kb[i] = kv[(long)n * (2 * NH * CH) + h * 128 + c];
    Vb[i] = kv[(long)n * (2 * NH * CH) + h * 128 + 64 + c];
}

// ---------------------------------------------------------------------------
// Build q/k/v points: res path = rot @ p + trans ; atom path = p + trans ; select.
// PR row: [qp_raw (3,48) | kvp_raw (3,144)] (GEMM natural (n,3*P) layout)
// PA row: [qpt_a (48,3)  | kvpt_a (144,3)]  (written with cns=3)
// ---------------------------------------------------------------------------
__global__ void k_points(const unsigned char* __restrict__ is_atom,
                         const float* __restrict__ rot, const float* __restrict__ trans,
                         const float* __restrict__ PR, const float* __restrict__ PA,
                         float* __restrict__ QPT, float* __restrict__ KPT,
                         float* __restrict__ VPT)
{
    int i = blockIdx.x * 256 + threadIdx.x;     // NRES * 192 points
    if (i >= NRES * 192) return;
    int n = i / 192, o = i % 192;
    const float* R = rot + n * 9;
    const float* t = trans + n * 3;
    bool at = is_atom[n] != 0;
    float out[3];
    if (o < 48) {                                // q points (h*4+pq)
        if (at) {
            for (int j = 0; j < 3; ++j) out[j] = PA[(long)n * 576 + o * 3 + j] + t[j];
        } else {
            float p0 = PR[(long)n * 576 + 0 * 48 + o];
            float p1 = PR[(long)n * 576 + 1 * 48 + o];
            float p2 = PR[(long)n * 576 + 2 * 48 + o];
            for (int r = 0; r < 3; ++r)
                out[r] = R[r * 3 + 0] * p0 + R[r * 3 + 1] * p1 + R[r * 3 + 2] * p2 + t[r];
        }
        for (int j = 0; j < 3; ++j) QPT[(long)n * 144 + o * 3 + j] = out[j];
    } else {                                     // kv points (h*12+slot)
        int ok = o - 48;
        if (at) {
            for (int j = 0; j < 3; ++j) out[j] = PA[(long)n * 576 + 144 + ok * 3 + j] + t[j];
        } else {
            float p0 = PR[(long)n * 576 + 144 + 0 * 144 + ok];
            float p1 = PR[(long)n * 576 + 144 + 1 * 144 + ok];
            float p2 = PR[(long)n * 576 + 144 + 2 * 144 + ok];
            for (int r = 0; r < 3; ++r)
                out[r] = R[r * 3 + 0] * p0 + R[r * 3 + 1] * p1 + R[r * 3 + 2] * p2 + t[r];
        }
        int h = ok / 12, pp = ok % 12;
        if (pp < PQ)
            for (int j = 0; j < 3; ++j) KPT[(long)n * 144 + (h * PQ + pp) * 3 + j] = out[j];
        else
            for (int j = 0; j < 3; ++j) VPT[(long)n * 288 + (h * PV + (pp - PQ)) * 3 + j] = out[j];
    }
}

// ---------------------------------------------------------------------------
// Fused epilogue: logits = qk*scale + sqrt(1/3)*b + pt_att + mask ; softmax over m.
// One block per (n, h) attention row.
// ---------------------------------------------------------------------------
__global__ void k_softmax(const float* __restrict__ QPT, const float* __restrict__ KPT,
                          const float* __restrict__ BPZ, const float* __restrict__ mask,
                          const float* __restrict__ head_weights, float* __restrict__ ATT)
{
    const int h = blockIdx.y;
    const int n = blockIdx.x;
    const int tid = threadIdx.x;

    __shared__ float qp[12];
    __shared__ float red[256];
    __shared__ float rowbuf[NRES];

    if (tid < 12) qp[tid] = QPT[(long)n * 144 + h * 12 + tid];
    __syncthreads();

    float hwv = head_weights[h];
    float sp  = (hwv > 20.f) ? hwv : log1pf(__expf(hwv));
    float hw  = sp * 0.13608276348795434f;          // sqrt(1/(3*(PQ*9/2))) = sqrt(1/54)
    float scale   = 0.07216878364870323f;           // sqrt(1/(3*C)) = sqrt(1/192)
    float sqrt13  = 0.57735026918962576f;           // sqrt(1/3)
    float maskn   = mask[n];

    float mx = -3.0e38f;
    float v3[3];
    for (int it = 0; it < 3; ++it) {
        int m = tid + it * 256;
        float v = ATT[((long)h * NRES + n) * NRES + m] * scale;
        v += sqrt13 * BPZ[((long)n * NRES + m) * 44 + h];
        const float* kp = KPT + (long)m * 144 + h * 12;
        float s2 = 0.f;
        for (int p = 0; p < PQ; ++p) {
            float dx = qp[p * 3 + 0] - kp[p * 3 + 0];
            float dy = qp[p * 3 + 1] - kp[p * 3 + 1];
            float dz = qp[p * 3 + 2] - kp[p * 3 + 2];
            s2 += dx * dx + dy * dy + dz * dz;
        }
        v += -0.5f * hw * s2;
        v += 100000.0f * (maskn * mask[m] - 1.0f);
        v3[it] = v;
        mx = fmaxf(mx, v);
    }
    red[tid] = mx; __syncthreads();
    for (int s = 128; s > 0; s >>= 1) {
        if (tid < s) red[tid] = fmaxf(red[tid], red[tid + s]);
        __syncthreads();
    }
    mx = red[0]; __syncthreads();

    float sum = 0.f;
    for (int it = 0; it < 3; ++it) {
        float e = __expf(v3[it] - mx);
        rowbuf[tid + it * 256] = e;
        sum += e;
    }
    red[tid] = sum; __syncthreads();
    for (int s = 128; s > 0; s >>= 1) {
        if (tid < s) red[tid] += red[tid + s];
        __syncthreads();
    }
    float inv = 1.f / red[0];
    for (int it = 0; it < 3; ++it) {
        int m = tid + it * 256;
        ATT[((long)h * NRES + n) * NRES + m] = rowbuf[m] * inv;
    }
}

// ---------------------------------------------------------------------------
// o_pt: local frame rotate (rot^T * (p - trans)), norm with EPS.
// ---------------------------------------------------------------------------
__global__ void k_finish_opt(const float* __restrict__ rot, const float* __restrict__ trans,
                             const float* __restrict__ OPTRAW,
                             float* __restrict__ OPT, float* __restrict__ OPTN)
{
    int i = blockIdx.x * 256 + threadIdx.x;     // NRES * 96
    if (i >= NRES * 96) return;
    int n = i / 96, p = i % 96;
    const float* R = rot + n * 9;
    const float* t = trans + n * 3;
    float g0 = OPTRAW[(long)n * 288 + p * 3 + 0] - t[0];
    float g1 = OPTRAW[(long)n * 288 + p * 3 + 1] - t[1];
    float g2 = OPTRAW[(long)n * 288 + p * 3 + 2] - t[2];
    float o0 = R[0] * g0 + R[3] * g1 + R[6] * g2;   // out_i = sum_j R[j][i] * g_j
    float o1 = R[1] * g0 + R[4] * g1 + R[7] * g2;
    float o2 = R[2] * g0 + R[5] * g1 + R[8] * g2;
    OPT[(long)n * 288 + p * 3 + 0] = o0;
    OPT[(long)n * 288 + p * 3 + 1] = o1;
    OPT[(long)n * 288 + p * 3 + 2] = o2;
    OPTN[(long)n * 96 + p] = sqrtf(o0 * o0 + o1 * o1 + o2 * o2 + 1e-8f);
}

// ---------------------------------------------------------------------------
// res_feats = concat[o | opt_x | opt_y | opt_z | opt_norm | o_pair]
// ---------------------------------------------------------------------------
__global__ void k_feats(const float* __restrict__ O, const float* __restrict__ OPT,
                        const float* __restrict__ OPTN, const float* __restrict__ OPAIR,
                        float* __restrict__ FEATS)
{
    long i = (long)blockIdx.x * 256 + threadIdx.x;  // NRES * 1536
    if (i >= (long)NRES * 1536) return;
    int n = (int)(i / 1536), j = (int)(i % 1536);
    float v;
    if (j < 768)       v = O[(long)n * 768 + j];
    else if (j < 1056) { int t = j - 768; int comp = t / 96; int p = t % 96;
                         v = OPT[(long)n * 288 + p * 3 + comp]; }
    else if (j < 1152) v = OPTN[(long)n * 96 + (j - 1056)];
    else               v = OPAIR[(long)n * 384 + (j - 1152)];
    FEATS[i] = v;
}

// ---------------------------------------------------------------------------
// Final select between atom-path and res-path outputs.
// ---------------------------------------------------------------------------
__global__ void k_final(const unsigned char* __restrict__ is_atom,
                        const float* __restrict__ RESOUT, const float* __restrict__ ASC,
                        const float* __restrict__ AVC, float* __restrict__ out)
{
    int i = blockIdx.x * 256 + threadIdx.x;     // NRES * 384
    if (i >= NRES * 384) return;
    int n = i / 384, j = i % 384;
    float v;
    if (is_atom[n])
        v = (j < 192) ? ASC[(long)n * 192 + j] : AVC[(long)n * 192 + (j - 192)];
    else
        v = RESOUT[i];
    out[i] = v;
}

// ---------------------------------------------------------------------------
// Host
// ---------------------------------------------------------------------------
static inline void gemm(hipStream_t st, const float* A, const float* B, const float* bias,
                        float* C, int M, int N, int K,
                        long ams, long aks, long bks, long bns, long cms, long cns,
                        int batches, long abat, long bbat, long cbat,
                        int beta, int waves_n)
{
    int wm = 8 / waves_n;                 // waves along M
    int bm = 16 * wm;                     // rows per block
    int bn = 32 * waves_n;                // cols per block (16x32 tile per wave)
    dim3 grid((M + bm - 1) / bm, (N + bn - 1) / bn, batches);
    bool ak1 = (aks == 1), bk1 = (bks == 1);
    if (ak1 && bk1)
        wmma_gemm_f32<true, true><<<grid, 256, 0, st>>>(A, B, bias, C, M, N, K,
            ams, aks, bks, bns, cms, cns, abat, bbat, cbat, beta, waves_n);
    else if (ak1)
        wmma_gemm_f32<true, false><<<grid, 256, 0, st>>>(A, B, bias, C, M, N, K,
            ams, aks, bks, bns, cms, cns, abat, bbat, cbat, beta, waves_n);
    else if (bk1)
        wmma_gemm_f32<false, true><<<grid, 256, 0, st>>>(A, B, bias, C, M, N, K,
            ams, aks, bks, bns, cms, cns, abat, bbat, cbat, beta, waves_n);
    else
        wmma_gemm_f32<false, false><<<grid, 256, 0, st>>>(A, B, bias, C, M, N, K,
            ams, aks, bks, bns, cms, cns, abat, bbat, cbat, beta, waves_n);
}

extern "C" void kernel_launch(void* const* d_in, const int* in_sizes, int n_in,
                              void* d_out, int out_size, void* d_ws, size_t ws_size,
                              hipStream_t stream)
{
    const float* s      = (const float*)d_in[0];
    const float* z      = (const float*)d_in[1];
    const float* rot    = (const float*)d_in[2];
    const float* trans  = (const float*)d_in[3];
    const float* mask   = (const float*)d_in[4];
    const unsigned char* is_atom = (const unsigned char*)d_in[5];
    const float* wq     = (const float*)d_in[6];
    const float* bq     = (const float*)d_in[7];
    const float* wkv    = (const float*)d_in[8];
    const float* bkv    = (const float*)d_in[9];
    const float* wqp    = (const float*)d_in[10];
    const float* bqp    = (const float*)d_in[11];
    const float* wkvp   = (const float*)d_in[12];
    const float* bkvp   = (const float*)d_in[13];
    const float* wb     = (const float*)d_in[14];
    const float* bb     = (const float*)d_in[15];
    const float* wdz    = (const float*)d_in[16];
    const float* bdz    = (const float*)d_in[17];
    const float* hweights = (const float*)d_in[18];
    const float* wout   = (const float*)d_in[19];
    const float* bout   = (const float*)d_in[20];
    const float* tfnq_ws  = (const float*)d_in[21];
    const float* tfnq_wv  = (const float*)d_in[22];
    const float* tfnkv_ws = (const float*)d_in[23];
    const float* tfnkv_wv = (const float*)d_in[24];
    const float* tfno_ws  = (const float*)d_in[25];
    const float* tfno_wv  = (const float*)d_in[26];
    float* out = (float*)d_out;

    const long NN = (long)NRES * NRES;
    float* W = (float*)d_ws;
    long off = 0;
    auto alloc = [&](long nf) { float* p = W + off; off += nf; return p; };
    float* Q      = alloc(NN);              // (n, H*C) final q
    float* Kb     = alloc(NN);              // (n, H*C) final k
    float* Vb     = alloc(NN);              // (n, H*C) final v
    float* QPT    = alloc(NRES * 144);      // (n, H*PQ, 3)
    float* KPT    = alloc(NRES * 144);
    float* VPT    = alloc(NRES * 288);      // (n, H*PV, 3)
    float* ATT    = alloc(12 * NN);         // (h, n, m) logits -> probs
    float* BPZ    = alloc(NN * 44);         // (n*m, [b(12)|pair_z(32)])
    float* WCAT   = alloc(128 * 44);
    float* BCAT   = alloc(64);
    float* O      = alloc(NN);              // scratch q_a, then (n, H*C) output o
    float* OPTRAW = alloc(NRES * 288);
    float* OPT    = alloc(NRES * 288);
    float* OPTN   = alloc(NRES * 96);
    float* OPAIR  = alloc(NRES * 384);
    float* FEATS  = alloc((long)NRES * 1536);
    float* RESOUT = alloc(NRES * 384);
    float* ASC    = alloc(NRES * 192);
    float* AVC    = alloc(NRES * 192);
    float* TMP_R  = alloc((long)NRES * 1536);   // kv_r raw
    float* TMP_A  = alloc((long)NRES * 1536);   // kv_a raw
    float* PPR    = alloc(NRES * 576);          // res point raw [qp(3,48)|kvp(3,144)]
    float* PPA    = alloc(NRES * 576);          // atom point raw [(48,3)|(144,3)]

    // 1) fused z projection weights, then z streamed ONCE: BPZ = z @ [wb|wdz] + [bb|bdz]
    k_concat_w<<<23, 256, 0, stream>>>(wb, bb, wdz, bdz, WCAT, BCAT);
    gemm(stream, z, WCAT, BCAT, BPZ, (int)NN, 44, CZ,
         CZ, 1, 44, 1, 44, 1, 1, 0, 0, 0, 0, 2);

    // 2) scalar projections (res + atom paths)
    gemm(stream, s, wq, bq, Q, NRES, NH * CH, CS, CS, 1, NH * CH, 1, NH * CH, 1,
         1, 0, 0, 0, 0, 4);
    gemm(stream, s, tfnq_ws, nullptr, O, NRES, NH * CH, N0C, CS, 1, NH * CH, 1,
         NH * CH, 1, 1, 0, 0, 0, 0, 4);
    gemm(stream, s, wkv, bkv, TMP_R, NRES, 2 * NH * CH, CS, CS, 1, 2 * NH * CH, 1,
         2 * NH * CH, 1, 1, 0, 0, 0, 0, 4);
    gemm(stream, s, tfnkv_ws, nullptr, TMP_A, NRES, 2 * NH * CH, N0C, CS, 1,
         2 * NH * CH, 1, 2 * NH * CH, 1, 1, 0, 0, 0, 0, 4);

    // 3) point projections (res path: (n, 3, P) layout; atom path: (n, P, 3) via cns=3)
    gemm(stream, s, wqp, bqp, PPR, NRES, 144, CS, CS, 1, 144, 1, 576, 1,
         1, 0, 0, 0, 0, 2);
    gemm(stream, s, wkvp, bkvp, PPR + 144, NRES, 432, CS, CS, 1, 432, 1, 576, 1,
         1, 0, 0, 0, 0, 2);
    for (int x = 0; x < 3; ++x) {
        gemm(stream, s + N0C + x, tfnq_wv, nullptr, PPA + x, NRES, 48, N1C,
             CS, 3, 48, 1, 576, 3, 1, 0, 0, 0, 0, 1);
        gemm(stream, s + N0C + x, tfnkv_wv, nullptr, PPA + 144 + x, NRES, 144, N1C,
             CS, 3, 144, 1, 576, 3, 1, 0, 0, 0, 0, 2);
    }

    // 4) selection + rigid transforms
    k_select_qkv<<<(int)((NN + 255) / 256), 256, 0, stream>>>(is_atom, O, TMP_R, TMP_A,
                                                              Q, Kb, Vb);
    k_points<<<(NRES * 192 + 255) / 256, 256, 0, stream>>>(is_atom, rot, trans,
                                                           PPR, PPA, QPT, KPT, VPT);

    // 5) attention logits qk^T (batched over heads); B has unit K-stride -> BK1 path
    gemm(stream, Q, Kb, nullptr, ATT, NRES, NRES, CH,
         NH * CH, 1,        /* A: q (n,c) */
         1, NH * CH,        /* B: k^T (c,m) from (m, h*64+c) */
         NRES, 1,           /* C: (n,m) */
         NH, CH, CH, NN, 0, 4);

    // 6) fused bias + point-attention + mask + softmax epilogue
    {
        dim3 g(NRES, NH);
        k_softmax<<<g, 256, 0, stream>>>(QPT, KPT, BPZ, mask, hweights, ATT);
    }

    // 7) o = a @ v (batched over heads)
    gemm(stream, ATT, Vb, nullptr, O, NRES, CH, NRES,
         NRES, 1, NH * CH, 1, NH * CH, 1,
         NH, NN, CH, CH, 0, 2);

    // 8) o_pt = a @ v_pts (batched over heads, N = PV*3 = 24)
    gemm(stream, ATT, VPT, nullptr, OPTRAW, NRES, PV * 3, NRES,
         NRES, 1, 288, 1, 288, 1,
         NH, NN, PV * 3, PV * 3, 0, 1);
    k_finish_opt<<<(NRES * 96 + 255) / 256, 256, 0, stream>>>(rot, trans, OPTRAW,
                                                              OPT, OPTN);

    // 9) o_pair = a @ pair_z (batched over n; pair_z = BPZ cols 12..43)
    gemm(stream, ATT, BPZ + NH, nullptr, OPAIR, NH, CZ4, NRES,
         NN, 1, 44, 1, CZ4, 1,
         NRES, NRES, (long)NRES * 44, 384, 0, 1);

    // 10) res path output
    k_feats<<<(int)(((long)NRES * 1536 + 255) / 256), 256, 0, stream>>>(O, OPT, OPTN,
                                                                        OPAIR, FEATS);
    gemm(stream, FEATS, wout, bout, RESOUT, NRES, CS, 1536,
         1536, 1, CS, 1, CS, 1, 1, 0, 0, 0, 0, 4);

    // 11) atom path scalar head: [o | opt_norm | o_pair] @ tfno_ws (split-K accumulate)
    gemm(stream, O, tfno_ws, nullptr, ASC, NRES, N0C, NH * CH,
         NH * CH, 1, N0C, 1, N0C, 1, 1, 0, 0, 0, 0, 2);
    gemm(stream, OPTN, tfno_ws + (long)(NH * CH) * N0C, nullptr, ASC, NRES, N0C, 96,
         96, 1, N0C, 1, N0C, 1, 1, 0, 0, 0, 1, 2);
    gemm(stream, OPAIR, tfno_ws + (long)(NH * CH + 96) * N0C, nullptr, ASC, NRES, N0C,
         384, 384, 1, N0C, 1, N0C, 1, 1, 0, 0, 0, 1, 2);

    // 12) atom path vector head: per-component GEMM, output interleaved (o,3)
    for (int x = 0; x < 3; ++x) {
        gemm(stream, OPT + x, tfno_wv, nullptr, AVC + x, NRES, N1C, NH * PV,
             288, 3, N1C, 1, 192, 3, 1, 0, 0, 0, 0, 2);
    }

    // 13) final select
    k_final<<<(NRES * 384 + 255) / 256, 256, 0, stream>>>(is_atom, RESOUT, ASC, AVC, out);
}